// GraphAttConv_73933567034040
// MI455X (gfx1250) — compile-verified
//
#include <hip/hip_runtime.h>
#include <hip/hip_bf16.h>
#include <math.h>

// GAT layer for MI455X (gfx1250, wave32).
// Phase 1: h = x @ W  via V_WMMA_F32_16X16X4_F32 (fp32 WMMA).
// Phase 2: per-node scores s1,s2 (decomposed attention dots).
// Phase 3: edge softmax (atomic max / sum) + weighted scatter-add, all L2-resident.

typedef __attribute__((ext_vector_type(2))) float v2f;
typedef __attribute__((ext_vector_type(8))) float v8f;

#define IN_F   256
#define HEADS  8
#define OUT_PH 16
#define CPN    128           // HEADS*OUT_PH, channels per node
#define NEG_SLOPE 0.2f
#define GAT_EPS 1e-16f

#define XT_PITCH 260         // 256 + 4 pad: conflict-free ds_load_b64 A-frag reads

// ---------------------------------------------------------------------------
// Kernel 1: WMMA GEMM.  h[n, hh*16+o] = sum_k x[n,k] * weight[hh,k,o]
// Block = 256 threads = 8 waves. Block owns 16 rows of x (staged in LDS),
// wave w computes the 16x16 tile for head w.  K-loop: 64 steps of 16x16x4.
// ---------------------------------------------------------------------------
__global__ void gat_gemm_wmma(const float* __restrict__ x,
                              const float* __restrict__ weight,
                              float* __restrict__ hbuf, int N) {
    extern __shared__ float xt[];                 // [16][XT_PITCH]
    const int tid     = threadIdx.x;
    const int rowBase = blockIdx.x * 16;

    // Cooperative, fully coalesced load of the 16x256 x-tile (rows >= N -> 0).
    #pragma unroll
    for (int i = 0; i < 16; ++i) {
        const int gr = rowBase + i;
        xt[i * XT_PITCH + tid] = (gr < N) ? x[(size_t)gr * IN_F + tid] : 0.0f;
    }
    __syncthreads();

    const int lane = tid & 31;
    const int hh   = tid >> 5;                    // wave id == head id
    const int l16  = lane & 15;                   // M index (A) / N index (B,D)
    const int half = lane >> 4;                   // K sub-pair selector

    const float* __restrict__ Wh = weight + (size_t)hh * IN_F * OUT_PH;

    v8f acc = {};
    #pragma unroll 8
    for (int kk = 0; kk < 64; ++kk) {
        const int k = kk * 4 + 2 * half;          // this lane's K pair
        // A fragment (16x4 f32): lane l16 row, K = k, k+1  -> one ds_load_b64
        v2f a = *reinterpret_cast<const v2f*>(&xt[l16 * XT_PITCH + k]);
        // B fragment (4x16 f32): column n = l16, rows K = k, k+1
        v2f b;
        b.x = Wh[(size_t)k * OUT_PH + l16];
        b.y = Wh[(size_t)(k + 1) * OUT_PH + l16];
        acc = __builtin_amdgcn_wmma_f32_16x16x4_f32(
            /*neg_a=*/false, a, /*neg_b=*/false, b,
            /*c_mod=*/(short)0, acc, /*reuse_a=*/false, /*reuse_b=*/false);
    }

    // D layout: lane = N (0..15 repeated), VGPR v -> M = v + 8*half
    #pragma unroll
    for (int v = 0; v < 8; ++v) {
        const int row = rowBase + v + 8 * half;
        if (row < N)
            hbuf[(size_t)row * CPN + hh * OUT_PH + l16] = acc[v];
    }
}

// ---------------------------------------------------------------------------
// Kernel 2: per-node attention scores + softmax-state init.
// s1[n,h] = dot(h[n,h,:], a[h,0:16]); s2[n,h] = dot(h[n,h,:], a[h,16:32])
// ---------------------------------------------------------------------------
__global__ void gat_node_scores(const float* __restrict__ hbuf,
                                const float* __restrict__ avec,
                                float* __restrict__ s1, float* __restrict__ s2,
                                float* __restrict__ mmax, float* __restrict__ denom,
                                int NH) {
    const int t = blockIdx.x * blockDim.x + threadIdx.x;
    if (t >= NH) return;
    const int n = t >> 3;
    const int h = t & 7;
    const float* __restrict__ hp = hbuf + (size_t)n * CPN + h * OUT_PH;
    const float* __restrict__ a1 = avec + h * 2 * OUT_PH;
    const float* __restrict__ a2 = a1 + OUT_PH;
    float d1 = 0.0f, d2 = 0.0f;
    #pragma unroll
    for (int i = 0; i < OUT_PH; ++i) {
        const float hv = hp[i];
        d1 = fmaf(hv, a1[i], d1);
        d2 = fmaf(hv, a2[i], d2);
    }
    s1[t] = d1;
    s2[t] = d2;
    mmax[t]  = -INFINITY;   // segment-max identity
    denom[t] = 0.0f;
}

__device__ __forceinline__ float leaky(float v) {
    return v > 0.0f ? v : NEG_SLOPE * v;
}

// Portable float atomic-max via int/uint ordering trick (init must be -inf).
__device__ __forceinline__ void atomicMaxFloat(float* addr, float value) {
    if (value >= 0.0f)
        atomicMax(reinterpret_cast<int*>(addr), __float_as_int(value));
    else
        atomicMin(reinterpret_cast<unsigned int*>(addr), __float_as_uint(value));
}

// ---------------------------------------------------------------------------
// Kernel 3: per-(edge, head) segment max of leaky(logit) over e0.
// ---------------------------------------------------------------------------
__global__ void gat_edge_max(const int* __restrict__ ei, int E,
                             const float* __restrict__ s1,
                             const float* __restrict__ s2,
                             float* __restrict__ mmax, int EH) {
    const int t = blockIdx.x * blockDim.x + threadIdx.x;
    if (t >= EH) return;
    const int e = t >> 3;
    const int h = t & 7;
    const int e0 = ei[e];
    const int e1 = ei[E + e];
    const float l = leaky(s1[e0 * HEADS + h] + s2[e1 * HEADS + h]);
    atomicMaxFloat(&mmax[e0 * HEADS + h], l);
}

// Kernel 3b: nodes with no out-edges: -inf -> 0 (matches reference isfinite fix)
__global__ void gat_fix_max(float* __restrict__ mmax, int NH) {
    const int t = blockIdx.x * blockDim.x + threadIdx.x;
    if (t >= NH) return;
    const float v = mmax[t];
    if (!isfinite(v)) mmax[t] = 0.0f;
}

// ---------------------------------------------------------------------------
// Kernel 4: denom[e0,h] += exp(logit - m[e0,h])
// ---------------------------------------------------------------------------
__global__ void gat_edge_denom(const int* __restrict__ ei, int E,
                               const float* __restrict__ s1,
                               const float* __restrict__ s2,
                               const float* __restrict__ mmax,
                               float* __restrict__ denom, int EH) {
    const int t = blockIdx.x * blockDim.x + threadIdx.x;
    if (t >= EH) return;
    const int e = t >> 3;
    const int h = t & 7;
    const int e0 = ei[e];
    const int e1 = ei[E + e];
    const int i0 = e0 * HEADS + h;
    const float l = leaky(s1[i0] + s2[e1 * HEADS + h]);
    atomicAdd(&denom[i0], __expf(l - mmax[i0]));
}

// ---------------------------------------------------------------------------
// Kernel 5: out[e0, c] += alpha[e,h] * h[e1, c].  128 threads per edge
// (c = h*16 + o); h[e1] read as 128 contiguous floats (coalesced, L2-hot).
// ---------------------------------------------------------------------------
__global__ void gat_edge_scatter(const int* __restrict__ ei, int E,
                                 const float* __restrict__ s1,
                                 const float* __restrict__ s2,
                                 const float* __restrict__ mmax,
                                 const float* __restrict__ denom,
                                 const float* __restrict__ hbuf,
                                 float* __restrict__ out, long long total) {
    const long long g = (long long)blockIdx.x * blockDim.x + threadIdx.x;
    if (g >= total) return;
    const int e = (int)(g >> 7);
    const int c = (int)(g & 127);
    const int h = c >> 4;
    const int e0 = ei[e];
    const int e1 = ei[E + e];
    const int i0 = e0 * HEADS + h;
    const float l     = leaky(s1[i0] + s2[e1 * HEADS + h]);
    const float alpha = __expf(l - mmax[i0]) / (denom[i0] + GAT_EPS);
    atomicAdd(&out[(size_t)e0 * CPN + c], alpha * hbuf[(size_t)e1 * CPN + c]);
}

// ---------------------------------------------------------------------------
extern "C" void kernel_launch(void* const* d_in, const int* in_sizes, int n_in,
                              void* d_out, int out_size, void* d_ws, size_t ws_size,
                              hipStream_t stream) {
    const float* x      = (const float*)d_in[0];
    const int*   ei     = (const int*)d_in[1];
    const float* weight = (const float*)d_in[2];
    const float* avec   = (const float*)d_in[3];
    float* out = (float*)d_out;

    const int N = in_sizes[0] / IN_F;
    const int E = in_sizes[1] / 2;
    const int NH = N * HEADS;
    const int EH = E * HEADS;

    // Workspace layout (floats): hbuf[N*128] | s1[N*8] | s2[N*8] | m[N*8] | denom[N*8]
    float* hbuf  = (float*)d_ws;
    float* s1    = hbuf  + (size_t)N * CPN;
    float* s2    = s1    + NH;
    float* mmax  = s2    + NH;
    float* denom = mmax  + NH;

    // 1) projection GEMM via fp32 WMMA
    {
        const int blocks = (N + 15) / 16;
        const size_t lds = 16 * XT_PITCH * sizeof(float);  // 16.25 KB
        gat_gemm_wmma<<<blocks, 256, lds, stream>>>(x, weight, hbuf, N);
    }
    // 2) per-node scores + softmax-state init
    gat_node_scores<<<(NH + 255) / 256, 256, 0, stream>>>(hbuf, avec, s1, s2,
                                                          mmax, denom, NH);
    // output accumulator init (d_out is poisoned by harness)
    hipMemsetAsync(out, 0, (size_t)N * CPN * sizeof(float), stream);

    // 3) segment max
    gat_edge_max<<<(EH + 255) / 256, 256, 0, stream>>>(ei, E, s1, s2, mmax, EH);
    gat_fix_max<<<(NH + 255) / 256, 256, 0, stream>>>(mmax, NH);
    // 4) softmax denominator
    gat_edge_denom<<<(EH + 255) / 256, 256, 0, stream>>>(ei, E, s1, s2, mmax,
                                                         denom, EH);
    // 5) weighted scatter-add
    {
        const long long total = (long long)E * CPN;
        const int blocks = (int)((total + 255) / 256);
        gat_edge_scatter<<<blocks, 256, 0, stream>>>(ei, E, s1, s2, mmax, denom,
                                                     hbuf, out, total);
    }
}